// GraphNet_55559696941202
// MI455X (gfx1250) — compile-verified
//
#include <hip/hip_runtime.h>
#include <cstdint>

typedef float v2f __attribute__((ext_vector_type(2)));
typedef float v8f __attribute__((ext_vector_type(8)));

#define CEILDIV(a, b) (((a) + (b) - 1) / (b))

// ---------------------------------------------------------------------------
// Zero the accumulator region of the workspace (deg + s128 + acc).
// ---------------------------------------------------------------------------
__global__ void k_zero(float* __restrict__ p, int n) {
  int i = blockIdx.x * blockDim.x + threadIdx.x;
  if (i < n) p[i] = 0.0f;
}

// ---------------------------------------------------------------------------
// Degree count over dst (self-loop added later as +1).
// ---------------------------------------------------------------------------
__global__ void k_deg(const int* __restrict__ dst, float* __restrict__ deg, int E) {
  int e = blockIdx.x * blockDim.x + threadIdx.x;
  if (e < E) atomicAdd(&deg[dst[e]], 1.0f);
}

// deg -> dinv = rsqrt(deg + 1)  (self-loop guarantees deg_total >= 1)
__global__ void k_dinv(float* __restrict__ deg, int N) {
  int n = blockIdx.x * blockDim.x + threadIdx.x;
  if (n < N) deg[n] = rsqrtf(deg[n] + 1.0f);
}

// ---------------------------------------------------------------------------
// g[n,:] = dinv[n] * emb[x[n], :]   (64 features, coalesced)
// ---------------------------------------------------------------------------
__global__ void k_gather(const int* __restrict__ x, const float* __restrict__ emb,
                         const float* __restrict__ dinv, float* __restrict__ g, int N) {
  int i = blockIdx.x * blockDim.x + threadIdx.x;
  if (i < N * 64) {
    int n = i >> 6, f = i & 63;
    g[i] = dinv[n] * emb[(long)x[n] * 64 + f];
  }
}

// ---------------------------------------------------------------------------
// Edge scatter: acc[d,:] += g[s,:].  One thread per (edge, feature); 64
// consecutive lanes share an edge -> coalesced loads and coalesced atomics.
// Working set (2 x 25.6MB) is L2-resident on MI455X (192MB L2).
// ---------------------------------------------------------------------------
__global__ void k_scatter(const int* __restrict__ src, const int* __restrict__ dst,
                          const float* __restrict__ g, float* __restrict__ acc, int E) {
  long i = (long)blockIdx.x * blockDim.x + threadIdx.x;
  long tot = (long)E * 64;
  if (i < tot) {
    int e = (int)(i >> 6), f = (int)(i & 63);
    int s = src[e], d = dst[e];
    atomicAdd(&acc[(long)d * 64 + f], g[(long)s * 64 + f]);
  }
}

// agg[n,:] = dinv[n] * (acc[n,:] + g[n,:])  (adds self-loop term; in place)
__global__ void k_finalize(float* __restrict__ acc, const float* __restrict__ g,
                           const float* __restrict__ dinv, int N) {
  int i = blockIdx.x * blockDim.x + threadIdx.x;
  if (i < N * 64) {
    int n = i >> 6;
    acc[i] = dinv[n] * (acc[i] + g[i]);
  }
}

// ---------------------------------------------------------------------------
// z = relu(agg @ W1 + b1); s128 = column-sum of z over all N rows.
// f32 WMMA 16x16x4, K=64 -> 16 WMMA per 16x16 output tile, 8 column tiles.
// W1 (64x128, 32KB) staged in LDS in B-fragment order.
// A 16x4 f32 fragment: lanes 0-15 row M=lane {K0,K1}; lanes 16-31 {K2,K3}.
// B 4x16 f32 fragment: mirrored; C/D: VGPR r = row r (lanes 0-15) / r+8.
// ---------------------------------------------------------------------------
__global__ __launch_bounds__(256) void k_gemm_relu_colsum(
    const float* __restrict__ agg, const float* __restrict__ W1,
    const float* __restrict__ b1, float* __restrict__ s128, int N) {
  __shared__ v2f ldsB[16 * 8 * 32];  // [kk][n][lane] -> 32 KB
  __shared__ float sred[128];

  const int t = threadIdx.x;
  // Stage W1 into LDS in fragment order.
  for (int e = t; e < 16 * 8 * 32; e += blockDim.x) {
    int lane = e & 31;
    int n = (e >> 5) & 7;
    int kk = e >> 8;
    int k0 = kk * 4 + ((lane & 16) ? 2 : 0);
    int col = n * 16 + (lane & 15);
    v2f v;
    v.x = W1[k0 * 128 + col];
    v.y = W1[(k0 + 1) * 128 + col];
    ldsB[e] = v;
  }
  if (t < 128) sred[t] = 0.0f;
  __syncthreads();

  const int lane = t & 31;
  const int lcol = lane & 15;
  const int half = lane >> 4;
  const int wid = blockIdx.x * (blockDim.x >> 5) + (t >> 5);
  const int nwaves = gridDim.x * (blockDim.x >> 5);
  const int ntiles = N >> 4;

  float bias[8];
#pragma unroll
  for (int n = 0; n < 8; ++n) bias[n] = b1[n * 16 + lcol];

  float colsum[8] = {0, 0, 0, 0, 0, 0, 0, 0};

  for (int tile = wid; tile < ntiles; tile += nwaves) {
    const float* arow = agg + (long)(tile * 16 + lcol) * 64 + half * 2;
    v2f a[16];
#pragma unroll
    for (int kk = 0; kk < 16; ++kk) a[kk] = *(const v2f*)(arow + kk * 4);

#pragma unroll
    for (int n = 0; n < 8; ++n) {
      v8f c = {};
#pragma unroll
      for (int kk = 0; kk < 16; ++kk) {
        c = __builtin_amdgcn_wmma_f32_16x16x4_f32(
            /*neg_a=*/false, a[kk], /*neg_b=*/false, ldsB[(kk * 8 + n) * 32 + lane],
            /*c_mod=*/(short)0, c, /*reuse_a=*/false, /*reuse_b=*/false);
      }
      float s = 0.0f;
#pragma unroll
      for (int r = 0; r < 8; ++r) s += fmaxf(c[r] + bias[n], 0.0f);
      colsum[n] += s;  // sum over this tile's 8 rows held by this lane
    }
  }

#pragma unroll
  for (int n = 0; n < 8; ++n) atomicAdd(&sred[n * 16 + lcol], colsum[n]);
  __syncthreads();
  if (t < 128) atomicAdd(&s128[t], sred[t]);

  // Tail rows if N % 16 != 0 (scalar fallback, block 0 / wave 0 only).
  if (blockIdx.x == 0 && t < 32) {
    for (int row = ntiles << 4; row < N; ++row) {
      for (int col = t; col < 128; col += 32) {
        float acc = b1[col];
        for (int k = 0; k < 64; ++k) acc += agg[(long)row * 64 + k] * W1[k * 128 + col];
        atomicAdd(&s128[col], fmaxf(acc, 0.0f));
      }
    }
  }
}

// ---------------------------------------------------------------------------
// out[j] = s128 . Wl[:,j] + N * bl[j]
// ---------------------------------------------------------------------------
__global__ void k_out(const float* __restrict__ s128, const float* __restrict__ Wl,
                      const float* __restrict__ bl, float* __restrict__ out, float Nf) {
  int j = threadIdx.x;
  if (j < 64) {
    float acc = Nf * bl[j];
    for (int k = 0; k < 128; ++k) acc += s128[k] * Wl[k * 64 + j];
    out[j] = acc;
  }
}

extern "C" void kernel_launch(void* const* d_in, const int* in_sizes, int n_in,
                              void* d_out, int out_size, void* d_ws, size_t ws_size,
                              hipStream_t stream) {
  const int* x = (const int*)d_in[0];       // (N,1) int
  const int* ei = (const int*)d_in[1];      // (2,E) int
  const float* emb = (const float*)d_in[2]; // (V,64)
  const float* W1 = (const float*)d_in[3];  // (64,128)
  const float* b1 = (const float*)d_in[4];  // (128)
  const float* Wl = (const float*)d_in[5];  // (128,64)
  const float* bl = (const float*)d_in[6];  // (64)
  float* out = (float*)d_out;               // (64)

  const int N = in_sizes[0];
  const int E = in_sizes[1] / 2;
  const int* src = ei;
  const int* dst = ei + E;

  // Workspace layout (floats): [deg/dinv: N][s128: 128][acc/agg: N*64][g: N*64]
  float* ws = (float*)d_ws;
  float* deg = ws;
  float* s128 = ws + N;
  float* acc = ws + (size_t)N + 128;
  float* g = ws + (size_t)N + 128 + (size_t)N * 64;

  const int B = 256;
  const int zn = N + 128 + N * 64;  // region that must start at zero

  k_zero<<<CEILDIV(zn, B), B, 0, stream>>>(ws, zn);
  k_deg<<<CEILDIV(E, B), B, 0, stream>>>(dst, deg, E);
  k_dinv<<<CEILDIV(N, B), B, 0, stream>>>(deg, N);
  k_gather<<<CEILDIV(N * 64, B), B, 0, stream>>>(x, emb, deg, g, N);
  long stot = (long)E * 64;
  k_scatter<<<(int)CEILDIV(stot, (long)B), B, 0, stream>>>(src, dst, g, acc, E);
  k_finalize<<<CEILDIV(N * 64, B), B, 0, stream>>>(acc, g, deg, N);
  k_gemm_relu_colsum<<<256, 256, 0, stream>>>(acc, W1, b1, s128, N);
  k_out<<<1, 64, 0, stream>>>(s128, Wl, bl, out, (float)N);
}